// GCNLayer_57982058496191
// MI455X (gfx1250) — compile-verified
//
#include <hip/hip_runtime.h>
#include <hip/hip_bf16.h>
#include <stdint.h>

// GCN layer: out[dst] += node_f[src] * rsqrt(out_deg[src]) * rsqrt(in_deg[dst])
// N=100000 nodes, d=32 features, E=1.6M edges. Entire working set (~39MB)
// is L2-resident (192MB L2) -> L2-atomic-bound scatter. WMMA not applicable
// (no dense matmul). CDNA5 paths used: TDM tensor_load_to_lds (double-buffered
// edge-index staging), s_wait_tensorcnt, wave32 lane mapping, f32 atomics.

#define FEAT 32
#define EPB  512   // edges per TDM tile (2KB of int32 per index array)
#define TPB  256   // 8 wave32 per block

#if defined(__HIP_DEVICE_COMPILE__) && defined(__gfx1250__) && __has_builtin(__builtin_amdgcn_tensor_load_to_lds)
#define USE_TDM 1
#else
#define USE_TDM 0
#endif

typedef unsigned int u32x4 __attribute__((ext_vector_type(4)));
typedef int          i32x8 __attribute__((ext_vector_type(8)));
typedef int          i32x4 __attribute__((ext_vector_type(4)));

#if USE_TDM
// 1-D TDM copy of int32 elements: global -> LDS. tensor_dim0 = elements
// remaining in the array so the tail of the tile zero-fills (TDM OOB reads
// return zero); compute loop still bounds-checks, zeros are just benign.
// 6-arg builtin form (clang-23 / therock-10.0): groups 0..3 + extra group + cpol;
// only groups 0 and 1 are meaningful for a <=2D tensor, rest are zero.
__device__ __forceinline__ void tdm_load_ints(unsigned lds_off, const int* gp,
                                              unsigned tileElems, unsigned remainElems)
{
    unsigned long long ga = (unsigned long long)(uintptr_t)gp;
    u32x4 g0;
    g0[0] = 1u;                                    // count = 1 (valid descriptor)
    g0[1] = lds_off;                               // lds_addr (bytes)
    g0[2] = (unsigned)ga;                          // global_addr[31:0]
    g0[3] = ((unsigned)(ga >> 32) & 0x01FFFFFFu)   // global_addr[56:32]
          | (2u << 30);                            // type = 2 ("image")
    i32x8 g1;
    g1[0] = (int)(2u << 16);                       // data_size = 4 bytes
    g1[1] = (int)((remainElems & 0xFFFFu) << 16);  // tensor_dim0[15:0]
    g1[2] = (int)(((remainElems >> 16) & 0xFFFFu)  // tensor_dim0[31:16]
          | (1u << 16));                           // tensor_dim1 = 1
    g1[3] = (int)(tileElems << 16);                // tile_dim0
    g1[4] = 1;                                     // tile_dim1 = 1, tile_dim2 = 0
    g1[5] = (int)remainElems;                      // tensor_dim0_stride lo (unused for 1 row)
    g1[6] = 0;                                     // stride hi / dim1_stride lo
    g1[7] = 0;                                     // dim1_stride hi
    i32x4 z4 = {0, 0, 0, 0};
    i32x8 z8 = {0, 0, 0, 0, 0, 0, 0, 0};
    __builtin_amdgcn_tensor_load_to_lds(g0, g1, z4, z4, z8, 0);
}
#endif

__device__ __forceinline__ int wave_id_scalar() {
#if defined(__HIP_DEVICE_COMPILE__)
    // force an SGPR compare -> scalar branch: only wave 0 issues the DMA
    return __builtin_amdgcn_readfirstlane((int)(threadIdx.x >> 5));
#else
    return (int)(threadIdx.x >> 5);
#endif
}

__device__ __forceinline__ void atomic_fadd(float* p, float v) {
#if defined(__HIP_DEVICE_COMPILE__)
    unsafeAtomicAdd(p, v);   // coarse-grained hipMalloc memory -> global_atomic_add_f32
#else
    (void)p; (void)v;
#endif
}

__global__ void gcn_zero_kernel(float* __restrict__ out, unsigned* __restrict__ cnt,
                                long long nOut, long long nCnt)
{
    long long i = (long long)blockIdx.x * blockDim.x + threadIdx.x;
    if (i < nOut) out[i] = 0.0f;
    if (i < nCnt) cnt[i] = 0u;
}

__global__ void gcn_degree_kernel(const int* __restrict__ src, const int* __restrict__ dst,
                                  unsigned* __restrict__ cntOut, unsigned* __restrict__ cntIn,
                                  int E)
{
    int e = blockIdx.x * blockDim.x + threadIdx.x;
    if (e < E) {
        atomicAdd(&cntOut[src[e]], 1u);
        atomicAdd(&cntIn[dst[e]], 1u);
    }
}

__global__ void gcn_rsqrt_kernel(unsigned* __restrict__ w, int n)
{
    int i = blockIdx.x * blockDim.x + threadIdx.x;
    if (i < n) {
        unsigned c = w[i];
        float deg = (float)(c < 1u ? 1u : c);   // clip(deg, 1)
        ((float*)w)[i] = rsqrtf(deg);           // in-place cnt -> rsqrt
    }
}

__global__ __launch_bounds__(TPB) void gcn_edge_agg_kernel(
    const float* __restrict__ node_f,
    const int*   __restrict__ src,
    const int*   __restrict__ dst,
    const float* __restrict__ rsOut,
    const float* __restrict__ rsIn,
    float*       __restrict__ out,
    int E, int numTiles)
{
    __shared__ int   sSrc[2][EPB];
    __shared__ int   sDst[2][EPB];
    __shared__ float sScale[EPB];

    const int tid  = threadIdx.x;
    int       tile = (int)blockIdx.x;      // grid <= numTiles: every block has work
    int       cur  = 0;

#if USE_TDM
    const bool wave0 = (wave_id_scalar() == 0);
    if (wave0) {
        const int base = tile * EPB;
        tdm_load_ints((unsigned)(uintptr_t)&sSrc[0][0], src + base, EPB, (unsigned)(E - base));
        tdm_load_ints((unsigned)(uintptr_t)&sDst[0][0], dst + base, EPB, (unsigned)(E - base));
    }
#endif

    for (; tile < numTiles; tile += (int)gridDim.x) {
        const int base = tile * EPB;

#if USE_TDM
        const int nxt = tile + (int)gridDim.x;
        if (wave0) {
            if (nxt < numTiles) {
                // prefetch next tile into the other buffer, then wait only for
                // the current tile's two DMAs (TENSORcnt completes in order)
                const int nb = nxt * EPB;
                tdm_load_ints((unsigned)(uintptr_t)&sSrc[cur ^ 1][0], src + nb, EPB, (unsigned)(E - nb));
                tdm_load_ints((unsigned)(uintptr_t)&sDst[cur ^ 1][0], dst + nb, EPB, (unsigned)(E - nb));
                __builtin_amdgcn_s_wait_tensorcnt(2);
            } else {
                __builtin_amdgcn_s_wait_tensorcnt(0);
            }
        }
        __syncthreads();   // current tile's LDS data visible to all 8 waves
#else
        for (int ei = tid; ei < EPB; ei += TPB) {
            const int e = base + ei;
            sSrc[cur][ei] = (e < E) ? src[e] : 0;
            sDst[cur][ei] = (e < E) ? dst[e] : 0;
        }
        __syncthreads();
#endif

        // phase 1: one normalization scale per edge (instead of 8x per lane)
        for (int ei = tid; ei < EPB; ei += TPB) {
            sScale[ei] = rsOut[sSrc[cur][ei]] * rsIn[sDst[cur][ei]];
        }
        __syncthreads();

        // phase 2: 8 lanes per edge (4 edges per wave32), float4 per lane ->
        // one B128 gather + 4 no-return f32 atomics per lane, full 128B row/edge
        const int lane = tid & 7;
        const int sub  = tid >> 3;                  // 0..31 edges per sweep
        for (int k = 0; k < EPB / (TPB / 8); ++k) { // 16 sweeps
            const int ei = sub + k * (TPB / 8);
            const int e  = base + ei;
            if (e < E) {
                const int   s  = sSrc[cur][ei];
                const int   d  = sDst[cur][ei];
                const float sc = sScale[ei];
                const float4 v = ((const float4*)(node_f + (size_t)s * FEAT))[lane];
                float* o = out + (size_t)d * FEAT + (size_t)(lane * 4);
                atomic_fadd(o + 0, v.x * sc);
                atomic_fadd(o + 1, v.y * sc);
                atomic_fadd(o + 2, v.z * sc);
                atomic_fadd(o + 3, v.w * sc);
            }
        }
        __syncthreads();   // everyone done reading buf[cur] before it is reused
        cur ^= 1;
    }
}

extern "C" void kernel_launch(void* const* d_in, const int* in_sizes, int n_in,
                              void* d_out, int out_size, void* d_ws, size_t ws_size,
                              hipStream_t stream)
{
    const float* node_f = (const float*)d_in[0];
    const int*   src    = (const int*)d_in[1];
    const int*   dst    = (const int*)d_in[2];
    float*       out    = (float*)d_out;

    const int N = in_sizes[0] / FEAT;   // 100000
    const int E = in_sizes[1];          // 1600000

    // ws layout: [0,N) out-degree counters, [N,2N) in-degree counters;
    // reused in place as float rsqrt tables after gcn_rsqrt_kernel. (800KB)
    unsigned* cntOut = (unsigned*)d_ws;
    unsigned* cntIn  = cntOut + N;

    const long long nOut = (long long)N * FEAT;
    const long long nCnt = 2LL * N;
    const long long nMax = (nOut > nCnt) ? nOut : nCnt;

    gcn_zero_kernel<<<(unsigned)((nMax + TPB - 1) / TPB), TPB, 0, stream>>>(out, cntOut, nOut, nCnt);
    gcn_degree_kernel<<<(unsigned)((E + TPB - 1) / TPB), TPB, 0, stream>>>(src, dst, cntOut, cntIn, E);
    gcn_rsqrt_kernel<<<(unsigned)((nCnt + TPB - 1) / TPB), TPB, 0, stream>>>(cntOut, (int)nCnt);

    const int numTiles = (E + EPB - 1) / EPB;            // 3125
    const int grid     = (numTiles < 1024) ? numTiles : 1024;  // persistent blocks -> TDM double-buffering
    gcn_edge_agg_kernel<<<(unsigned)grid, TPB, 0, stream>>>(
        node_f, src, dst, (const float*)cntOut, (const float*)cntIn, out, E, numTiles);
}